// QNNEmbedding_9560597201464
// MI455X (gfx1250) — compile-verified
//
#include <hip/hip_runtime.h>

// ---------------------------------------------------------------------------
// 12-qubit QNN state-vector simulator for MI455X (gfx1250, wave32).
//
//  * 1 workgroup (256 threads / 8 waves) per batch element; complex f32 state
//    (4096 amplitudes) lives in LDS as separate real/imag planes (32 KB).
//  * Per layer: fuse the 4 rotations per wire into one 2x2 complex gate,
//    Kron-fuse groups of 4 wires into three 16x16 complex unitaries, then
//    apply each as 16 M-tiles of a complex GEMM using V_WMMA_F32_16X16X4_F32
//    (Dr = Sr*Br + Si*(-Bi), Di = Sr*Bi + Si*Br, with B = U^T).
//  * The CNOT ladder composes to the binary->Gray permutation x ^= (x>>1);
//    folded into LDS addressing as g^layer. Both g^layer and the tile
//    addressing are GF(2)-linear, so all 16 addresses per tile are derived
//    from per-stage bases with single XORs (minimal VALU).
//  * Stage + tile loops fully unrolled: no control flow in the WMMA region,
//    EXEC stays all-ones (WMMA requirement).
// ---------------------------------------------------------------------------

#define QNN_NWIRES  12
#define QNN_DIM     4096
#define QNN_LAYERS  4
#define QNN_OUT     8
#define QNN_BLOCK   256
#define QNN_WAVES   (QNN_BLOCK / 32)          // 8
#define QNN_TILES   (16 / QNN_WAVES)          // 2 M-tiles per wave

typedef __attribute__((ext_vector_type(2))) float v2f;
typedef __attribute__((ext_vector_type(8))) float v8f;

struct c2 { float r, i; };
struct M2 { c2 e[2][2]; };

__device__ __forceinline__ c2 cmul(c2 a, c2 b) {
  c2 c; c.r = a.r * b.r - a.i * b.i; c.i = a.r * b.i + a.i * b.r; return c;
}
__device__ __forceinline__ c2 cadd(c2 a, c2 b) {
  c2 c; c.r = a.r + b.r; c.i = a.i + b.i; return c;
}
__device__ __forceinline__ M2 mmul(const M2& A, const M2& B) {
  M2 C;
  #pragma unroll
  for (int i = 0; i < 2; ++i)
    #pragma unroll
    for (int j = 0; j < 2; ++j)
      C.e[i][j] = cadd(cmul(A.e[i][0], B.e[0][j]), cmul(A.e[i][1], B.e[1][j]));
  return C;
}
__device__ __forceinline__ M2 mRX(float t) {
  float c = __cosf(0.5f * t), s = __sinf(0.5f * t);
  M2 M;
  M.e[0][0] = {c, 0.f};  M.e[0][1] = {0.f, -s};
  M.e[1][0] = {0.f, -s}; M.e[1][1] = {c, 0.f};
  return M;
}
__device__ __forceinline__ M2 mRY(float t) {
  float c = __cosf(0.5f * t), s = __sinf(0.5f * t);
  M2 M;
  M.e[0][0] = {c, 0.f}; M.e[0][1] = {-s, 0.f};
  M.e[1][0] = {s, 0.f}; M.e[1][1] = {c, 0.f};
  return M;
}
__device__ __forceinline__ M2 mRZ(float t) {
  float c = __cosf(0.5f * t), s = __sinf(0.5f * t);
  M2 M;
  M.e[0][0] = {c, -s};    M.e[0][1] = {0.f, 0.f};
  M.e[1][0] = {0.f, 0.f}; M.e[1][1] = {c, s};
  return M;
}

// logical state index from (8-bit rest index mg, group nibble k); STAGE is a
// compile-time constant -> pure (disjoint) bit placement, GF(2)-linear.
template <int STAGE>
__device__ __forceinline__ int stageAddr(int mg, int k) {
  if (STAGE == 0) return (k << 8) | mg;                               // wires 0-3  (bits 11:8)
  if (STAGE == 1) return ((mg & 0xF0) << 4) | (k << 4) | (mg & 0x0F); // wires 4-7  (bits 7:4)
  return (mg << 4) | k;                                               // wires 8-11 (bits 3:0)
}

__global__ __launch_bounds__(QNN_BLOCK)
void qnn_statevec_kernel(const float* __restrict__ xin,
                         const float* __restrict__ win,
                         float* __restrict__ out) {
  __shared__ float sR[QNN_DIM];
  __shared__ float sI[QNN_DIM];
  __shared__ float uR[3 * 256];
  __shared__ float uI[3 * 256];
  __shared__ float gR[QNN_NWIRES * 4];
  __shared__ float gI[QNN_NWIRES * 4];
  __shared__ float red[QNN_OUT * QNN_BLOCK];

  const int tid  = threadIdx.x;
  const int b    = blockIdx.x;
  const int lane = tid & 31;
  const int wave = tid >> 5;       // 0..7
  const int n    = lane & 15;      // M index for A, N index for B/D
  const int kh   = lane >> 4;      // K-half selector (A/B layout)

  const float x0 = xin[2 * b + 0];
  const float x1 = xin[2 * b + 1];

  // |0...0>
  for (int p = tid; p < QNN_DIM; p += QNN_BLOCK) { sR[p] = 0.f; sI[p] = 0.f; }
  if (tid == 0) sR[0] = 1.f;
  __syncthreads();

  for (int layer = 0; layer < QNN_LAYERS; ++layer) {
    // Branchless Gray-power masks (uniform per layer): g^l state permutation.
    const int m1 = -(layer & 1);
    const int m2 = -((layer >> 1) & 1);

    // ---- fused per-wire 2x2 gate: M = RZ(w2) RY(w1) RX(w0) Enc(x) ----
    if (tid < QNN_NWIRES) {
      const int wire = tid;
      M2 M = (wire & 1) ? mRY(x1) : mRX(x0);
      const float* wv = win + (layer * QNN_NWIRES + wire) * 3;
      M = mmul(mRX(wv[0]), M);
      M = mmul(mRY(wv[1]), M);
      M = mmul(mRZ(wv[2]), M);
      #pragma unroll
      for (int i = 0; i < 2; ++i)
        #pragma unroll
        for (int j = 0; j < 2; ++j) {
          gR[wire * 4 + i * 2 + j] = M.e[i][j].r;
          gI[wire * 4 + i * 2 + j] = M.e[i][j].i;
        }
    }
    __syncthreads();

    // ---- Kron-fuse 4 wires -> 16x16 group unitary U[g'][g] ----
    #pragma unroll
    for (int rep = 0; rep < 3; ++rep) {               // 768 entries / 256 thr
      const int idx = rep * QNN_BLOCK + tid;
      const int grp = idx >> 8;
      const int e   = idx & 255;
      const int row = e >> 4;   // g' (wire 4g at nibble bit3)
      const int col = e & 15;   // g
      c2 v = {1.f, 0.f};
      #pragma unroll
      for (int q = 0; q < 4; ++q) {
        const int i = (row >> (3 - q)) & 1;
        const int j = (col >> (3 - q)) & 1;
        const int w = grp * 4 + q;
        c2 g = { gR[w * 4 + i * 2 + j], gI[w * 4 + i * 2 + j] };
        v = cmul(v, g);
      }
      uR[grp * 256 + row * 16 + col] = v.r;
      uI[grp * 256 + row * 16 + col] = v.i;
    }
    __syncthreads();

    // ---- three complex GEMM stages, fully unrolled (no control flow) ----
    auto gray = [&](int x) {                // branchless g^layer (GF(2)-linear)
      x ^= (x >> 1) & m1;
      x ^= (x >> 2) & m2;
      return x;
    };

    auto runStage = [&](auto stage_c) {
      constexpr int STAGE = decltype(stage_c)::value;

      // B = U^T in WMMA B-layout: B[k][n] = U[n][k]; pre-negated imag copy.
      v2f Br[4], Bi[4], Bni[4];
      #pragma unroll
      for (int c = 0; c < 4; ++c) {
        const int k0 = 4 * c + 2 * kh;
        const int u0 = STAGE * 256 + n * 16 + k0;
        Br[c][0]  =  uR[u0];  Br[c][1]  =  uR[u0 + 1];
        Bi[c][0]  =  uI[u0];  Bi[c][1]  =  uI[u0 + 1];
        Bni[c][0] = -uI[u0];  Bni[c][1] = -uI[u0 + 1];
      }

      // Addresses are gray(stageAddr(.)) of disjoint bit fields; both maps
      // are GF(2)-linear, so factor them into per-stage bases + XOR deltas.
      const int baseA = gray(stageAddr<STAGE>(wave * 16 + n, 0));       // per lane
      const int baseD = gray(stageAddr<STAGE>(wave * 16 + 8 * kh, n));  // per lane
      const int dTT   = gray(stageAddr<STAGE>(QNN_WAVES * 16, 0));      // tile delta
      const int dKp1  = gray(stageAddr<STAGE>(0, 1));                   // K+1 delta
      int dK[4];
      #pragma unroll
      for (int c = 0; c < 4; ++c)
        dK[c] = gray(stageAddr<STAGE>(0, 4 * c + 2 * kh));
      int dV[8];                                                        // uniform
      #pragma unroll
      for (int v = 0; v < 8; ++v)
        dV[v] = gray(stageAddr<STAGE>(v, 0));

      // 2 M-tiles per wave; each tile's read set == write set (bijection),
      // so tiles are independent and no intra-stage barrier is needed.
      #pragma unroll
      for (int tt = 0; tt < QNN_TILES; ++tt) {
        const int tBaseA = tt ? (baseA ^ dTT) : baseA;
        const int tBaseD = tt ? (baseD ^ dTT) : baseD;

        v2f Ar[4], Ai[4];
        #pragma unroll
        for (int c = 0; c < 4; ++c) {
          const int p0 = tBaseA ^ dK[c];
          const int p1 = p0 ^ dKp1;
          Ar[c][0] = sR[p0]; Ar[c][1] = sR[p1];
          Ai[c][0] = sI[p0]; Ai[c][1] = sI[p1];
        }

        v8f accR = {};
        v8f accI = {};
        #pragma unroll
        for (int c = 0; c < 4; ++c) {
          accR = __builtin_amdgcn_wmma_f32_16x16x4_f32(
              false, Ar[c], false, Br[c], (short)0, accR, false, false);
          accI = __builtin_amdgcn_wmma_f32_16x16x4_f32(
              false, Ar[c], false, Bi[c], (short)0, accI, false, false);
        }
        #pragma unroll
        for (int c = 0; c < 4; ++c) {
          accR = __builtin_amdgcn_wmma_f32_16x16x4_f32(
              false, Ai[c], false, Bni[c], (short)0, accR, false, false);
          accI = __builtin_amdgcn_wmma_f32_16x16x4_f32(
              false, Ai[c], false, Br[c], (short)0, accI, false, false);
        }

        // D layout: lane holds N = lane&15; VGPR v -> M = v + 8*(lane>=16)
        #pragma unroll
        for (int v = 0; v < 8; ++v) {
          const int p = tBaseD ^ dV[v];
          sR[p] = accR[v];
          sI[p] = accI[v];
        }
      }
      __syncthreads();
    };

    runStage(std::integral_constant<int, 0>{});
    runStage(std::integral_constant<int, 1>{});
    runStage(std::integral_constant<int, 2>{});
  }

  // ---- readout: expval_i = pi * sum_x sign_i(x) * |amp(x)|^2 ----
  float acc[QNN_OUT];
  #pragma unroll
  for (int i = 0; i < QNN_OUT; ++i) acc[i] = 0.f;
  #pragma unroll 4
  for (int rep = 0; rep < QNN_DIM / QNN_BLOCK; ++rep) {
    const int x = rep * QNN_BLOCK + tid;
    const int p = (x ^ (x >> 4)) & (QNN_DIM - 1);    // g^4: final Gray map
    const float ar = sR[p], ai = sI[p];
    const float pr = ar * ar + ai * ai;
    #pragma unroll
    for (int i = 0; i < QNN_OUT; ++i)
      acc[i] += ((x >> (11 - i)) & 1) ? -pr : pr;
  }
  #pragma unroll
  for (int i = 0; i < QNN_OUT; ++i) red[i * QNN_BLOCK + tid] = acc[i];
  __syncthreads();
  if (tid < QNN_OUT) {
    float s = 0.f;
    for (int j = 0; j < QNN_BLOCK; ++j) s += red[tid * QNN_BLOCK + j];
    out[b * QNN_OUT + tid] = s * 3.14159265358979323846f;
  }
}

extern "C" void kernel_launch(void* const* d_in, const int* in_sizes, int n_in,
                              void* d_out, int out_size, void* d_ws, size_t ws_size,
                              hipStream_t stream) {
  const float* x = (const float*)d_in[0];   // [B, 2] f32
  const float* w = (const float*)d_in[1];   // [4, 12, 3] f32
  float* out = (float*)d_out;               // [B, 8] f32
  const int batch = in_sizes[0] / 2;
  qnn_statevec_kernel<<<batch, QNN_BLOCK, 0, stream>>>(x, w, out);
}